// Net_38165079392910
// MI455X (gfx1250) — compile-verified
//
#include <hip/hip_runtime.h>

typedef __attribute__((ext_vector_type(2))) float v2f;
typedef __attribute__((ext_vector_type(8))) float v8f;

#define FDIM 32

// ---------------- degree / normalization ----------------

__global__ void k_deg_init(float* __restrict__ deg, int n) {
  int i = blockIdx.x * blockDim.x + threadIdx.x;
  if (i < n) deg[i] = 1.0f;  // self-loop contributes 1 to every node's degree
}

__global__ void k_deg_count(const long long* __restrict__ dst, float* __restrict__ deg, int e) {
  int i = blockIdx.x * blockDim.x + threadIdx.x;
  if (i < e) atomicAdd(&deg[(int)dst[i]], 1.0f);
}

__global__ void k_dis(float* __restrict__ deg, int n) {
  int i = blockIdx.x * blockDim.x + threadIdx.x;
  if (i < n) deg[i] = rsqrtf(deg[i]);  // deg >= 1 always (self loops)
}

// ---------------- fc1: [N,2] @ [2,32] + b, relu ----------------

__global__ void k_fc1_relu(const float* __restrict__ x, const float* __restrict__ w,
                           const float* __restrict__ b, float* __restrict__ h, int n) {
  int i = blockIdx.x * blockDim.x + threadIdx.x;
  if (i >= n * FDIM) return;
  int node = i >> 5, j = i & 31;
  float v = fmaf(x[node * 2 + 1], w[FDIM + j], fmaf(x[node * 2], w[j], b[j]));
  h[i] = fmaxf(v, 0.0f);
}

// ---------------- GEMM: [N,32] @ [32,32] via V_WMMA_F32_16X16X4_F32 ----------------
// One wave computes a 16-row x 32-col tile: two 16x16 f32 accumulators, 8 chained
// K=4 steps => 16 v_wmma_f32_16x16x4_f32 per wave. EXEC is all-1s at every WMMA
// (the only early exit is wave-uniform). Epilogue is fused with the GCN self-loop
// term: writes hw (for the edge scatter) AND acc = hw * dis^2 (scatter accumulator
// seed), saving a full 19.2MB read pass per layer.

__global__ void k_gemm32_wmma(const float* __restrict__ h, const float* __restrict__ W,
                              const float* __restrict__ dis,
                              float* __restrict__ hw, float* __restrict__ acc, int n) {
  int lane = threadIdx.x & 31;
  int wave = blockIdx.x * (blockDim.x >> 5) + (threadIdx.x >> 5);
  int row0 = wave * 16;
  if (row0 >= n) return;                 // uniform across the wave
  int lo = lane & 15;
  int hi = lane >> 4;                    // 0: K=0,1 half; 1: K=2,3 half
  int row = row0 + lo;
  if (row >= n) row = n - 1;             // clamp (n % 16 == 0 in practice)

  v8f acc0 = {};                         // cols 0..15
  v8f acc1 = {};                         // cols 16..31
#pragma unroll
  for (int k = 0; k < 8; ++k) {
    int kk = 4 * k + 2 * hi;
    // A fragment (16x4): lane(lo,hi) holds A[M=lo][K=kk], A[M=lo][K=kk+1]
    v2f a;
    a.x = h[row * FDIM + kk];
    a.y = h[row * FDIM + kk + 1];
    // B fragments (4x16): lane(lo,hi) holds B[K=kk][N=lo(+16)], B[K=kk+1][N=lo(+16)]
    v2f b0, b1;
    b0.x = W[kk * FDIM + lo];
    b0.y = W[(kk + 1) * FDIM + lo];
    b1.x = W[kk * FDIM + 16 + lo];
    b1.y = W[(kk + 1) * FDIM + 16 + lo];
    acc0 = __builtin_amdgcn_wmma_f32_16x16x4_f32(false, a, false, b0, (short)0, acc0, false, false);
    acc1 = __builtin_amdgcn_wmma_f32_16x16x4_f32(false, a, false, b1, (short)0, acc1, false, false);
  }
  // C/D layout: VGPR r -> row (r + 8*hi), col lo (+16 for acc1)
  if (row0 + 16 <= n) {                  // wave-uniform full-tile fast path
#pragma unroll
    for (int r = 0; r < 8; ++r) {
      int orow = row0 + r + 8 * hi;
      float dn = dis[orow];
      float dn2 = dn * dn;
      float v0 = acc0[r], v1 = acc1[r];
      hw[orow * FDIM + lo]       = v0;
      hw[orow * FDIM + 16 + lo]  = v1;
      acc[orow * FDIM + lo]      = v0 * dn2;
      acc[orow * FDIM + 16 + lo] = v1 * dn2;
    }
  } else {                               // guarded tail (not hit for N=150000)
#pragma unroll
    for (int r = 0; r < 8; ++r) {
      int orow = row0 + r + 8 * hi;
      if (orow < n) {
        float dn = dis[orow];
        float dn2 = dn * dn;
        float v0 = acc0[r], v1 = acc1[r];
        hw[orow * FDIM + lo]       = v0;
        hw[orow * FDIM + 16 + lo]  = v1;
        acc[orow * FDIM + lo]      = v0 * dn2;
        acc[orow * FDIM + 16 + lo] = v1 * dn2;
      }
    }
  }
}

// ---------------- edge scatter: acc[dst] += hw[src] * dis[src]*dis[dst] ----------------
// One wave per edge (lane == feature); index loads broadcast from L0, payload rows
// are 128B contiguous per edge and L2-resident (hw is 19.2 MB << 192 MB L2).

__global__ void k_scatter(const long long* __restrict__ src, const long long* __restrict__ dst,
                          const float* __restrict__ hw, const float* __restrict__ dis,
                          float* __restrict__ acc, int e) {
  int i = blockIdx.x * blockDim.x + threadIdx.x;
  if (i >= e * FDIM) return;             // 76.8M < 2^31
  int eid = i >> 5, j = i & 31;
  int s  = (int)src[eid];
  int d0 = (int)dst[eid];
  float v = hw[s * FDIM + j] * (dis[s] * dis[d0]);
  atomicAdd(&acc[d0 * FDIM + j], v);
}

// ---------------- bias + relu (float4 vectorized: b128 loads/stores) ----------------

__global__ void k_bias_relu(const float4* __restrict__ acc, const float* __restrict__ b,
                            float4* __restrict__ out, int n) {
  int i = blockIdx.x * blockDim.x + threadIdx.x;   // over n * 8 float4s
  if (i >= n * (FDIM / 4)) return;
  int jb = (i & 7) * 4;                            // feature base within row
  float4 v = acc[i];
  v.x = fmaxf(v.x + b[jb + 0], 0.0f);
  v.y = fmaxf(v.y + b[jb + 1], 0.0f);
  v.z = fmaxf(v.z + b[jb + 2], 0.0f);
  v.w = fmaxf(v.w + b[jb + 3], 0.0f);
  out[i] = v;
}

// ---------------- head: [N,32] @ [32,1] + b -> interleaved output column ----------------

__global__ void k_head(const float4* __restrict__ g, const float* __restrict__ w,
                       const float* __restrict__ b, float* __restrict__ out, int n, int col) {
  int node = blockIdx.x * blockDim.x + threadIdx.x;
  if (node >= n) return;
  float s = b[0];
#pragma unroll
  for (int q = 0; q < FDIM / 4; ++q) {
    float4 v = g[node * (FDIM / 4) + q];
    s = fmaf(v.x, w[q * 4 + 0], s);
    s = fmaf(v.y, w[q * 4 + 1], s);
    s = fmaf(v.z, w[q * 4 + 2], s);
    s = fmaf(v.w, w[q * 4 + 3], s);
  }
  out[node * 2 + col] = s;
}

// ---------------- driver ----------------

extern "C" void kernel_launch(void* const* d_in, const int* in_sizes, int n_in,
                              void* d_out, int out_size, void* d_ws, size_t ws_size,
                              hipStream_t stream) {
  const float*     x      = (const float*)d_in[0];
  const long long* ei     = (const long long*)d_in[1];   // int64 [2, E]
  const float*     fc1_w  = (const float*)d_in[2];
  const float*     fc1_b  = (const float*)d_in[3];
  const float*     w1     = (const float*)d_in[4];
  const float*     b1     = (const float*)d_in[5];
  const float*     w2     = (const float*)d_in[6];
  const float*     b2     = (const float*)d_in[7];
  const float*     w31    = (const float*)d_in[8];
  const float*     b31    = (const float*)d_in[9];
  const float*     w32    = (const float*)d_in[10];
  const float*     b32    = (const float*)d_in[11];
  const float*     fc21_w = (const float*)d_in[12];
  const float*     fc21_b = (const float*)d_in[13];
  const float*     fc22_w = (const float*)d_in[14];
  const float*     fc22_b = (const float*)d_in[15];
  float* out = (float*)d_out;

  const int n = in_sizes[0] / 2;    // x is [N, 2]
  const int e = in_sizes[1] / 2;    // edge_index is [2, E]
  const long long* src = ei;
  const long long* dst = ei + e;

  // workspace: dis[n] | B0[n*32] | B1[n*32] | B2[n*32]
  size_t nf = (size_t)n * FDIM;
  float* dis = (float*)d_ws;
  size_t off = ((size_t)n * sizeof(float) + 255) & ~(size_t)255;
  float* B0 = (float*)((char*)d_ws + off);
  float* B1 = B0 + nf;
  float* B2 = B1 + nf;

  const int T = 256;
  int blkN  = (n + T - 1) / T;
  int blkNF = (int)((nf + T - 1) / T);
  int blkN4 = (int)((nf / 4 + T - 1) / T);
  int blkE  = (e + T - 1) / T;
  int blkEF = (int)(((size_t)e * FDIM + T - 1) / T);
  int waves = (n + 15) / 16;
  int blkG  = (waves + 7) / 8;      // 8 waves per 256-thread block

  // --- symmetric normalization coefficients (from dst + self loops) ---
  k_deg_init <<<blkN,  T, 0, stream>>>(dis, n);
  k_deg_count<<<blkE,  T, 0, stream>>>(dst, dis, e);
  k_dis      <<<blkN,  T, 0, stream>>>(dis, n);

  // --- fc1 + relu: x -> B0 ---
  k_fc1_relu <<<blkNF, T, 0, stream>>>(x, fc1_w, fc1_b, B0, n);

  // --- conv1: B0 -> B0 ---
  k_gemm32_wmma<<<blkG,  T, 0, stream>>>(B0, w1, dis, B1, B2, n);
  k_scatter    <<<blkEF, T, 0, stream>>>(src, dst, B1, dis, B2, e);
  k_bias_relu  <<<blkN4, T, 0, stream>>>((const float4*)B2, b1, (float4*)B0, n);

  // --- conv2: B0 -> B0 ---
  k_gemm32_wmma<<<blkG,  T, 0, stream>>>(B0, w2, dis, B1, B2, n);
  k_scatter    <<<blkEF, T, 0, stream>>>(src, dst, B1, dis, B2, e);
  k_bias_relu  <<<blkN4, T, 0, stream>>>((const float4*)B2, b2, (float4*)B0, n);

  // --- branch 1: conv31 (B0 preserved) -> head -> out[:,0] ---
  k_gemm32_wmma<<<blkG,  T, 0, stream>>>(B0, w31, dis, B1, B2, n);
  k_scatter    <<<blkEF, T, 0, stream>>>(src, dst, B1, dis, B2, e);
  k_bias_relu  <<<blkN4, T, 0, stream>>>((const float4*)B2, b31, (float4*)B1, n);
  k_head       <<<blkN,  T, 0, stream>>>((const float4*)B1, fc21_w, fc21_b, out, n, 0);

  // --- branch 2: conv32 -> head -> out[:,1] ---
  k_gemm32_wmma<<<blkG,  T, 0, stream>>>(B0, w32, dis, B1, B2, n);
  k_scatter    <<<blkEF, T, 0, stream>>>(src, dst, B1, dis, B2, e);
  k_bias_relu  <<<blkN4, T, 0, stream>>>((const float4*)B2, b32, (float4*)B1, n);
  k_head       <<<blkN,  T, 0, stream>>>((const float4*)B1, fc22_w, fc22_b, out, n, 1);
}